// latent_encoder_gelu_6983616823975
// MI455X (gfx1250) — compile-verified
//
#include <hip/hip_runtime.h>
#include <math.h>

// CDNA5 (gfx1250) implementation. Matrix work goes through the native fp32
// WMMA pipe: V_WMMA_F32_16X16X4_F32 (wave32, 16x16 D tiles). The im2col B
// operand is staged in LDS (320KB/WGP) so the hot loop is ds_load + wmma.

typedef __attribute__((ext_vector_type(2))) float v2f;
typedef __attribute__((ext_vector_type(8))) float v8f;

#define HH 64
#define WW 64
#define MAXC 96   // max input channels across convs (conv1: 96, others: 64)

__device__ __forceinline__ float gelu_exact(float x) {
    return 0.5f * x * (1.0f + erff(x * 0.70710678118654752f));
}

// ---------------- pixel unshuffle (4x): [16,6,256,256] -> [16,96,64,64] ----------
__global__ void k_pixel_unshuffle(const float* __restrict__ in, float* __restrict__ out, int total) {
    int i = blockIdx.x * blockDim.x + threadIdx.x;
    if (i >= total) return;
    int w = i & 63; int t = i >> 6;
    int h = t & 63; t >>= 6;
    int oc = t % 96; int b = t / 96;
    int c = oc >> 4; int r = oc & 15; int ki = r >> 2; int kj = r & 3;
    out[i] = in[(((size_t)b * 6 + c) * 256 + h * 4 + ki) * 256 + w * 4 + kj];
}

// ---------------- per-(b,c) spatial mean -> pooled[b*C+c] ------------------------
__global__ void __launch_bounds__(256) k_pool_mean(const float* __restrict__ x,
                                                   float* __restrict__ pooled, int C) {
    int bc = blockIdx.x;
    const float* p = x + (size_t)bc * (HH * WW);
    float s = 0.f;
    for (int i = threadIdx.x; i < HH * WW; i += 256) s += p[i];
    __shared__ float red[256];
    red[threadIdx.x] = s; __syncthreads();
    for (int off = 128; off > 0; off >>= 1) {
        if (threadIdx.x < off) red[threadIdx.x] += red[threadIdx.x + off];
        __syncthreads();
    }
    if (threadIdx.x == 0) pooled[bc] = red[0] * (1.0f / (HH * WW));
}

// ---------------- h = relu(pooled @ w1^T + b1) -----------------------------------
__global__ void k_mlp_h(const float* __restrict__ pooled, const float* __restrict__ w1,
                        const float* __restrict__ b1, float* __restrict__ h, int C, int mid) {
    int b = blockIdx.x; int m = threadIdx.x;
    if (m >= mid) return;
    const float* pr = pooled + b * C;
    const float* wr = w1 + m * C;
    float s = b1[m];
    for (int c = 0; c < C; ++c) s += pr[c] * wr[c];
    h[b * mid + m] = fmaxf(s, 0.f);
}

// ---------------- logits = h @ w2^T + b2 -----------------------------------------
__global__ void k_mlp_logits(const float* __restrict__ h, const float* __restrict__ w2,
                             const float* __restrict__ b2, float* __restrict__ kern,
                             int mid, int OCK, int total) {
    int i = blockIdx.x * blockDim.x + threadIdx.x;
    if (i >= total) return;
    int j = i % OCK; int b = i / OCK;
    const float* hr = h + b * mid;
    const float* wr = w2 + (size_t)j * mid;
    float s = b2[j];
    for (int m = 0; m < mid; ++m) s += hr[m] * wr[m];
    kern[(size_t)b * OCK + j] = s;
}

// ---------------- softmax over each row of length OCK ----------------------------
__global__ void __launch_bounds__(256) k_softmax_row(float* __restrict__ kern, int OCK) {
    int b = blockIdx.x;
    float* row = kern + (size_t)b * OCK;
    __shared__ float red[256];
    int t = threadIdx.x;
    float mx = -1e30f;
    for (int i = t; i < OCK; i += 256) mx = fmaxf(mx, row[i]);
    red[t] = mx; __syncthreads();
    for (int off = 128; off > 0; off >>= 1) {
        if (t < off) red[t] = fmaxf(red[t], red[t + off]);
        __syncthreads();
    }
    mx = red[0]; __syncthreads();
    float s = 0.f;
    for (int i = t; i < OCK; i += 256) s += expf(row[i] - mx);
    red[t] = s; __syncthreads();
    for (int off = 128; off > 0; off >>= 1) {
        if (t < off) red[t] += red[t + off];
        __syncthreads();
    }
    float inv = 1.0f / red[0];
    for (int i = t; i < OCK; i += 256) row[i] = expf(row[i] - mx) * inv;
}

// ---------------- per-sample dyn conv as implicit GEMM via WMMA ------------------
// D[64, 4096] = Kern[64, C*9] x Patches[C*9, 4096]. One wave owns a 64x16
// output slab. The input halo tile (C x 3 x 18) is staged in LDS once; the
// K-loop is then ds_load (B) + b64 global loads (A) + 4x wmma.
// mode: 0 = plain, 1 = gelu, 2 = + residual
__global__ void __launch_bounds__(32)
k_dynconv_wmma(const float* __restrict__ x, const float* __restrict__ kern,
               const float* res, float* out, int C, int mode) {
    const int Kdim = C * 9;
    const int b    = blockIdx.y;
    const int tile = blockIdx.x;            // 0..255 : h * (W/16) + wtile
    const int h    = tile >> 2;
    const int w0   = (tile & 3) << 4;
    const int lane = threadIdx.x;
    const int n    = lane & 15;
    const int half = lane >> 4;             // 0: K base, 1: K+2 base
    const int wn   = w0 + n;

    const float* kb_ = kern + (size_t)b * (64 * (size_t)Kdim);
    const float* xb  = x + (size_t)b * C * (HH * WW);

    // --- stage halo tile: patch[c][ki][dw], dw in [0,18) maps to ww = w0+dw-1 ---
    __shared__ float patch[MAXC * 3 * 18];
    const int fillN = C * 54;               // 5184 (C=96) or 3456 (C=64): /32 exact
    for (int idx = lane; idx < fillN; idx += 32) {
        int c  = idx / 54;
        int r  = idx - 54 * c;
        int ki = r / 18;
        int dw = r - 18 * ki;
        int hh = h + ki - 1;
        int ww = w0 + dw - 1;
        float v = 0.f;
        if ((unsigned)hh < (unsigned)HH && (unsigned)ww < (unsigned)WW)
            v = xb[((size_t)c * HH + hh) * WW + ww];
        patch[idx] = v;
    }
    __syncthreads();

    v8f acc0 = {0.f,0.f,0.f,0.f,0.f,0.f,0.f,0.f};
    v8f acc1 = acc0, acc2 = acc0, acc3 = acc0;

    #pragma unroll 2
    for (int k = 0; k < Kdim; k += 4) {
        const int kb = k + half * 2;
        // B fragment from LDS: kidx -> (c, ki, kj); ww = wn+kj-1 -> dw = n+kj
        v2f bf;
        #pragma unroll
        for (int q = 0; q < 2; ++q) {
            int kidx = kb + q;
            int c  = kidx / 9;
            int r  = kidx - 9 * c;
            int ki = r / 3;
            int kj = r - 3 * ki;
            bf[q] = patch[c * 54 + ki * 18 + n + kj];
        }
        // A fragments (row m = n) for the 4 output-channel tiles: aligned b64 loads
        const float* arow = kb_ + (size_t)n * Kdim + kb;
        __builtin_prefetch(arow + 8, 0, 0);     // global_prefetch_b8
        v2f a0 = *(const v2f*)(arow);
        v2f a1 = *(const v2f*)(arow + (size_t)16 * Kdim);
        v2f a2 = *(const v2f*)(arow + (size_t)32 * Kdim);
        v2f a3 = *(const v2f*)(arow + (size_t)48 * Kdim);

        acc0 = __builtin_amdgcn_wmma_f32_16x16x4_f32(false, a0, false, bf, (short)0, acc0, false, false);
        acc1 = __builtin_amdgcn_wmma_f32_16x16x4_f32(false, a1, false, bf, (short)0, acc1, false, false);
        acc2 = __builtin_amdgcn_wmma_f32_16x16x4_f32(false, a2, false, bf, (short)0, acc2, false, false);
        acc3 = __builtin_amdgcn_wmma_f32_16x16x4_f32(false, a3, false, bf, (short)0, acc3, false, false);
    }

    // Epilogue: C/D layout -> VGPR v holds M=v (lanes 0-15) / M=v+8 (lanes 16-31)
    #pragma unroll
    for (int v = 0; v < 8; ++v) {
        int m = v + half * 8;
        float vals[4] = { acc0[v], acc1[v], acc2[v], acc3[v] };
        #pragma unroll
        for (int ot = 0; ot < 4; ++ot) {
            int o = ot * 16 + m;
            size_t idx = (((size_t)b * 64 + o) * HH + h) * WW + wn;
            float val = vals[ot];
            if (mode == 1)      val = gelu_exact(val);
            else if (mode == 2) val += res[idx];
            out[idx] = val;
        }
    }
}

// ---------------- 4x4 adaptive pool + reshape to [b, 16 tokens, 64 ch] ----------
__global__ void k_pool44(const float* __restrict__ x, float* __restrict__ xseq, int total) {
    int i = blockIdx.x * blockDim.x + threadIdx.x;
    if (i >= total) return;
    int g = i & 15; int t = i >> 4;
    int c = t & 63; int b = t >> 6;
    int gh = g >> 2, gw = g & 3;
    const float* p = x + (((size_t)b * 64 + c) * HH + gh * 16) * WW + gw * 16;
    float s = 0.f;
    for (int ii = 0; ii < 16; ++ii)
        for (int jj = 0; jj < 16; ++jj) s += p[ii * WW + jj];
    xseq[((size_t)b * 16 + g) * 64 + c] = s * (1.0f / 256.0f);
}

// ---------------- token-mixing MLP block (one block per sample) ------------------
__global__ void __launch_bounds__(256) k_mixer_token(float* xseq,
        const float* __restrict__ g1,  const float* __restrict__ be1,
        const float* __restrict__ pm1w, const float* __restrict__ pm1b,
        const float* __restrict__ pm2w, const float* __restrict__ pm2b) {
    int b = blockIdx.x;
    float* xb = xseq + (size_t)b * 1024;
    __shared__ float xs[1024], ys[1024], t1s[64 * 8], mu[16], rsig[16];
    int t = threadIdx.x;
    for (int i = t; i < 1024; i += 256) xs[i] = xb[i];
    __syncthreads();
    if (t < 16) {
        float m = 0.f;
        for (int c = 0; c < 64; ++c) m += xs[t * 64 + c];
        m *= (1.f / 64.f);
        float v = 0.f;
        for (int c = 0; c < 64; ++c) { float d = xs[t * 64 + c] - m; v += d * d; }
        v *= (1.f / 64.f);
        mu[t] = m; rsig[t] = rsqrtf(v + 1e-5f);
    }
    __syncthreads();
    for (int i = t; i < 1024; i += 256) {
        int p = i >> 6, c = i & 63;
        ys[i] = (xs[i] - mu[p]) * rsig[p] * g1[c] + be1[c];
    }
    __syncthreads();
    for (int i = t; i < 512; i += 256) {          // [c=64, j=8]
        int c = i >> 3, j = i & 7;
        float s = pm1b[j];
        for (int p = 0; p < 16; ++p) s += ys[p * 64 + c] * pm1w[j * 16 + p];
        t1s[i] = gelu_exact(s);
    }
    __syncthreads();
    for (int i = t; i < 1024; i += 256) {         // [p=16, c=64]
        int p = i >> 6, c = i & 63;
        float s = pm2b[p];
        for (int j = 0; j < 8; ++j) s += t1s[c * 8 + j] * pm2w[p * 8 + j];
        xb[i] = xs[i] + s;
    }
}

// ---------------- channel-mixing MLP block (one block per sample) ----------------
__global__ void __launch_bounds__(256) k_mixer_channel(float* xseq,
        const float* __restrict__ g2,  const float* __restrict__ be2,
        const float* __restrict__ cm1w, const float* __restrict__ cm1b,
        const float* __restrict__ cm2w, const float* __restrict__ cm2b) {
    int b = blockIdx.x;
    float* xb = xseq + (size_t)b * 1024;
    __shared__ float xs[1024], us[1024], hbuf[16 * 256], mu[16], rsig[16];
    int t = threadIdx.x;
    for (int i = t; i < 1024; i += 256) xs[i] = xb[i];
    __syncthreads();
    if (t < 16) {
        float m = 0.f;
        for (int c = 0; c < 64; ++c) m += xs[t * 64 + c];
        m *= (1.f / 64.f);
        float v = 0.f;
        for (int c = 0; c < 64; ++c) { float d = xs[t * 64 + c] - m; v += d * d; }
        v *= (1.f / 64.f);
        mu[t] = m; rsig[t] = rsqrtf(v + 1e-5f);
    }
    __syncthreads();
    for (int i = t; i < 1024; i += 256) {
        int p = i >> 6, c = i & 63;
        us[i] = (xs[i] - mu[p]) * rsig[p] * g2[c] + be2[c];
    }
    __syncthreads();
    for (int i = t; i < 4096; i += 256) {         // [p=16, k=256]
        int p = i >> 8, k = i & 255;
        float s = cm1b[k];
        const float* wr = cm1w + (size_t)k * 64;
        for (int c = 0; c < 64; ++c) s += us[p * 64 + c] * wr[c];
        hbuf[i] = gelu_exact(s);
    }
    __syncthreads();
    for (int i = t; i < 1024; i += 256) {         // [p=16, c=64]
        int p = i >> 6, c = i & 63;
        float s = cm2b[c];
        const float* wr = cm2w + (size_t)c * 256;
        const float* hr = hbuf + p * 256;
        for (int k = 0; k < 256; ++k) s += hr[k] * wr[k];
        xb[i] = xs[i] + s;
    }
}

// ---------------- head: out = gelu(xseq @ endw^T + endb) -> [16,16,256] ----------
__global__ void k_end(const float* __restrict__ xseq, const float* __restrict__ ew,
                      const float* __restrict__ eb, float* __restrict__ out, int total) {
    int i = blockIdx.x * blockDim.x + threadIdx.x;
    if (i >= total) return;
    int k = i & 255; int t = i >> 8;
    int p = t & 15; int b = t >> 4;
    const float* xr = xseq + ((size_t)b * 16 + p) * 64;
    const float* wr = ew + (size_t)k * 64;
    float s = eb[k];
    for (int c = 0; c < 64; ++c) s += xr[c] * wr[c];
    out[i] = gelu_exact(s);
}

// ---------------- host-side sequencing ------------------------------------------
static void run_dynconv(hipStream_t stream, const float* xin, int C,
                        const float* w1, const float* b1, const float* w2, const float* b2,
                        int mid, float* pooled, float* hbuf, float* kern,
                        const float* res, float* out, int mode) {
    const int B = 16;
    const int OCK = 64 * C * 9;
    k_pool_mean<<<dim3(B * C), dim3(256), 0, stream>>>(xin, pooled, C);
    k_mlp_h<<<dim3(B), dim3(32), 0, stream>>>(pooled, w1, b1, hbuf, C, mid);
    int total = B * OCK;
    k_mlp_logits<<<dim3((total + 255) / 256), dim3(256), 0, stream>>>(hbuf, w2, b2, kern, mid, OCK, total);
    k_softmax_row<<<dim3(B), dim3(256), 0, stream>>>(kern, OCK);
    k_dynconv_wmma<<<dim3(256, 16), dim3(32), 0, stream>>>(xin, kern, res, out, C, mode);
}

extern "C" void kernel_launch(void* const* d_in, const int* in_sizes, int n_in,
                              void* d_out, int out_size, void* d_ws, size_t ws_size,
                              hipStream_t stream) {
    // Input order (setup_inputs dict, depth-first):
    // 0: inp_img
    // 1-4:   conv1 {w1,b1,w2,b2}  (mid=24, C=96)
    // 5+8i:  blocks[i].c1 {w1,b1,w2,b2}, 9+8i: blocks[i].c2 {...}  (mid=16, C=64)
    // 37-40: conv2 {w1,b1,w2,b2}
    // 41-44: mlp {g1,be1,g2,be2}
    // 45-48: pm1{w,b}, pm2{w,b}
    // 49-52: cm1{w,b}, cm2{w,b}
    // 53-54: end{w,b}
    const float* inp = (const float*)d_in[0];
    auto P = [&](int i) { return (const float*)d_in[i]; };

    char* ws = (char*)d_ws;
    size_t off = 0;
    auto alloc = [&](size_t bytes) -> void* {
        void* p = ws + off;
        off += (bytes + 255) & ~(size_t)255;
        return p;
    };
    float* x0     = (float*)alloc(16ull * 96 * 64 * 64 * 4);
    float* xP     = (float*)alloc(16ull * 64 * 64 * 64 * 4);
    float* xQ     = (float*)alloc(16ull * 64 * 64 * 64 * 4);
    float* pooled = (float*)alloc(16ull * 96 * 4);
    float* hbuf   = (float*)alloc(16ull * 24 * 4);
    float* kern   = (float*)alloc(16ull * 55296 * 4);
    float* xseq   = (float*)alloc(16ull * 16 * 64 * 4);
    (void)ws_size; (void)in_sizes; (void)n_in; (void)out_size;

    int tot0 = 16 * 96 * 64 * 64;
    k_pixel_unshuffle<<<(tot0 + 255) / 256, 256, 0, stream>>>(inp, x0, tot0);

    // conv1 + gelu
    run_dynconv(stream, x0, 96, P(1), P(2), P(3), P(4), 24, pooled, hbuf, kern, nullptr, xP, 1);

    // residual blocks
    for (int i = 0; i < 4; ++i) {
        int base = 5 + i * 8;
        run_dynconv(stream, xP, 64, P(base + 0), P(base + 1), P(base + 2), P(base + 3), 16,
                    pooled, hbuf, kern, nullptr, xQ, 1);                       // gelu(c1(x))
        run_dynconv(stream, xQ, 64, P(base + 4), P(base + 5), P(base + 6), P(base + 7), 16,
                    pooled, hbuf, kern, xP, xP, 2);                            // c2(..) + x
    }

    // conv2 (plain)
    run_dynconv(stream, xP, 64, P(37), P(38), P(39), P(40), 16, pooled, hbuf, kern, nullptr, xQ, 0);

    // 4x4 pool -> tokens
    int tp = 16 * 64 * 16;
    k_pool44<<<(tp + 255) / 256, 256, 0, stream>>>(xQ, xseq, tp);

    // mixer
    k_mixer_token<<<16, 256, 0, stream>>>(xseq, P(41), P(42), P(45), P(46), P(47), P(48));
    k_mixer_channel<<<16, 256, 0, stream>>>(xseq, P(43), P(44), P(49), P(50), P(51), P(52));

    // head
    int te = 16 * 16 * 256;
    k_end<<<(te + 255) / 256, 256, 0, stream>>>(xseq, P(53), P(54), (float*)d_out, te);
}